// Model_87900800680533
// MI455X (gfx1250) — compile-verified
//
#include <hip/hip_runtime.h>
#include <cstddef>

// ---------------- model dimensions ----------------
#define BZ   2
#define SL   20
#define CH   6
#define TT   30
#define DD   200
#define NSEQ (BZ*SL*CH)   // 240 sequences in the trunk
#define RR   (NSEQ*TT)    // 7200 rows
#define DIN  400          // D_INNER
#define DST  16           // D_STATE
#define DTR  13           // DT_RANK

typedef float v2f __attribute__((ext_vector_type(2)));
typedef float v8f __attribute__((ext_vector_type(8)));
// 4-byte-aligned vector load type (rows may start at odd-float offsets, e.g. lda=45)
typedef float v2f_a4 __attribute__((ext_vector_type(2), aligned(4)));

__device__ __forceinline__ float gelu_f(float x) {
  return 0.5f * x * (1.0f + erff(x * 0.70710678118654752f));
}
__device__ __forceinline__ float softplus_f(float x) {
  return fmaxf(x, 0.0f) + log1pf(__expf(-fabsf(x)));
}
__device__ __forceinline__ float silu_f(float x) {
  return x / (1.0f + __expf(-x));
}

// ---------------------------------------------------------------------------
// Generic WMMA GEMM:  C[M,N] = act(A[M,K] @ W[N,K]^T + bias[N]) + res[M,N]
// One wave computes one 16x16 tile with V_WMMA_F32_16X16X4_F32 (fp32 in/out).
//
// Bounds handling without exec-mask churn:
//  * Edge rows/cols: clamp the row pointer to a valid row and load
//    unconditionally. Garbage A-row m only affects C-row m; garbage B-col n
//    only affects C-col n; both are masked at the store. No zeroing needed.
//  * K tail (K % 4 != 0): load from a clamped index into a temp, then select
//    zero with a VALU cndmask (select on the value, not the load).
// ACT: 0=none, 1=gelu, 2=softplus.  RES: add residual after activation.
// ---------------------------------------------------------------------------
template<int ACT, bool RES>
__global__ void gemm_wmma_kernel(const float* __restrict__ A, int lda,
                                 const float* __restrict__ W, int ldw,
                                 const float* __restrict__ bias,
                                 const float* __restrict__ res, int ldres,
                                 float* __restrict__ C, int ldc,
                                 int M, int N, int K)
{
  const int tiles_n = (N + 15) >> 4;
  const int tiles_m = (M + 15) >> 4;
  const int wave = blockIdx.x * (blockDim.x >> 5) + (threadIdx.x >> 5);
  if (wave >= tiles_m * tiles_n) return;            // wave-uniform branch
  const int tm = wave / tiles_n;
  const int tn = wave - tm * tiles_n;
  const int m0 = tm << 4, n0 = tn << 4;
  const int lane = threadIdx.x & 31;
  const int hf = lane >> 4;                          // 0 or 1 (half-wave)
  const int lr = lane & 15;
  const int arow = m0 + lr;
  const int brow = n0 + lr;
  // clamp to a valid row; invalid lanes compute garbage that is masked at store
  const int arowc = arow < M ? arow : (M - 1);
  const int browc = brow < N ? brow : (N - 1);
  const float* Ap = A + (size_t)arowc * lda + 2 * hf;
  const float* Wp = W + (size_t)browc * ldw + 2 * hf;

  v8f c = {0.f, 0.f, 0.f, 0.f, 0.f, 0.f, 0.f, 0.f};
  int k = 0;
  for (; k + 4 <= K; k += 4) {
    v2f av = (v2f)*(const v2f_a4*)(Ap + k);          // global_load_b64
    v2f bv = (v2f)*(const v2f_a4*)(Wp + k);
    c = __builtin_amdgcn_wmma_f32_16x16x4_f32(false, av, false, bv,
                                              (short)0, c, false, false);
  }
  if (k < K) {                                       // K tail (e.g. K=13)
    const int k0 = k + 2 * hf, k1 = k0 + 1;
    const int k0c = k0 < K ? k0 : (K - 1);
    const int k1c = k1 < K ? k1 : (K - 1);
    const float* Ab = A + (size_t)arowc * lda;
    const float* Wb = W + (size_t)browc * ldw;
    const float a0 = Ab[k0c], a1 = Ab[k1c];          // unconditional loads
    const float b0 = Wb[k0c], b1 = Wb[k1c];
    v2f av, bv;
    av.x = (k0 < K) ? a0 : 0.f;                      // v_cndmask, no exec flips
    av.y = (k1 < K) ? a1 : 0.f;
    bv.x = (k0 < K) ? b0 : 0.f;
    bv.y = (k1 < K) ? b1 : 0.f;
    c = __builtin_amdgcn_wmma_f32_16x16x4_f32(false, av, false, bv,
                                              (short)0, c, false, false);
  }

  // C/D layout: VGPR i -> (row m0+i, col n0+lane) lanes 0-15; rows +8 lanes 16-31
  const int col = n0 + lr;
  if (col < N) {
    const float bvs = bias ? bias[col] : 0.f;
#pragma unroll
    for (int i = 0; i < 8; ++i) {
      const int row = m0 + i + 8 * hf;
      if (row < M) {
        float v = c[i] + bvs;
        if (ACT == 1)      v = gelu_f(v);
        else if (ACT == 2) v = softplus_f(v);
        if (RES)           v += res[(size_t)row * ldres + col];
        C[(size_t)row * ldc + col] = v;
      }
    }
  }
}

// ------------------------- norms (one wave per row) -------------------------
__global__ void rmsnorm_kernel(const float* __restrict__ x,
                               const float* __restrict__ w,
                               float* __restrict__ o, int rows, int D)
{
  const int row = blockIdx.x * (blockDim.x >> 5) + (threadIdx.x >> 5);
  if (row >= rows) return;
  const int lane = threadIdx.x & 31;
  const float* xp = x + (size_t)row * D;
  float ss = 0.f;
  for (int i = lane; i < D; i += 32) { float v = xp[i]; ss += v * v; }
  for (int off = 16; off; off >>= 1) ss += __shfl_xor(ss, off, 32);
  const float inv = rsqrtf(ss / (float)D + 1e-5f);
  float* op = o + (size_t)row * D;
  for (int i = lane; i < D; i += 32) op[i] = xp[i] * w[i] * inv;
}

__global__ void layernorm_kernel(const float* __restrict__ x,
                                 const float* __restrict__ w,
                                 const float* __restrict__ b,
                                 float* __restrict__ o, int rows, int D)
{
  const int row = blockIdx.x * (blockDim.x >> 5) + (threadIdx.x >> 5);
  if (row >= rows) return;
  const int lane = threadIdx.x & 31;
  const float* xp = x + (size_t)row * D;
  float s = 0.f;
  for (int i = lane; i < D; i += 32) s += xp[i];
  for (int off = 16; off; off >>= 1) s += __shfl_xor(s, off, 32);
  const float mean = s / (float)D;
  float v = 0.f;
  for (int i = lane; i < D; i += 32) { float t = xp[i] - mean; v += t * t; }
  for (int off = 16; off; off >>= 1) v += __shfl_xor(v, off, 32);
  const float inv = rsqrtf(v / (float)D + 1e-5f);
  float* op = o + (size_t)row * D;
  for (int i = lane; i < D; i += 32) op[i] = (xp[i] - mean) * inv * w[i] + b[i];
}

// ------------- Mamba depthwise causal conv (k=4) + SiLU ---------------------
// xz: [NSEQ*L, 2*DIN]; xm is the first DIN columns. xc: [NSEQ*L, DIN].
__global__ void mamba_conv_kernel(const float* __restrict__ xz,
                                  const float* __restrict__ cW,  // [DIN,4]
                                  const float* __restrict__ cb,  // [DIN]
                                  float* __restrict__ xc, int nseq, int L)
{
  const int idx = blockIdx.x * blockDim.x + threadIdx.x;
  const int total = nseq * L * DIN;
  if (idx >= total) return;
  const int d = idx % DIN;
  const int t = (idx / DIN) % L;
  const int n = idx / (DIN * L);
  float acc = cb[d];
#pragma unroll
  for (int kk = 0; kk < 4; ++kk) {
    const int tt = t + kk - 3;                      // pad (3,0) on the left
    if (tt >= 0) acc += cW[d * 4 + kk] * xz[((size_t)(n * L + tt)) * (2 * DIN) + d];
  }
  xc[idx] = silu_f(acc);
}

// ----------------- Mamba selective scan (one thread per (n,d)) --------------
// y[n,t,d] = (scan(s) . C + D[d]*xc) * silu(z[n,t,d])
__global__ void mamba_scan_kernel(const float* __restrict__ xc,    // [nseq,L,DIN]
                                  const float* __restrict__ delta, // [nseq,L,DIN]
                                  const float* __restrict__ dbl,   // [nseq,L,45] B@13 C@29
                                  const float* __restrict__ Alog,  // [DIN,DST]
                                  const float* __restrict__ Dp,    // [DIN]
                                  const float* __restrict__ xz,    // z at col DIN, ld 2*DIN
                                  float* __restrict__ y,           // [nseq,L,DIN]
                                  int nseq, int L)
{
  const int idx = blockIdx.x * blockDim.x + threadIdx.x;
  if (idx >= nseq * DIN) return;
  const int d = idx % DIN;
  const int n = idx / DIN;
  float A[DST], st[DST];
#pragma unroll
  for (int s = 0; s < DST; ++s) { A[s] = -__expf(Alog[d * DST + s]); st[s] = 0.f; }
  const float Dd = Dp[d];
  for (int t = 0; t < L; ++t) {
    const size_t r = (size_t)(n * L + t);
    const float dl  = delta[r * DIN + d];
    const float xcv = xc[r * DIN + d];
    const float* Bm = dbl + r * (DTR + 2 * DST) + DTR;
    const float* Cm = Bm + DST;
    float acc = 0.f;
#pragma unroll
    for (int s = 0; s < DST; ++s) {
      st[s] = __expf(dl * A[s]) * st[s] + dl * Bm[s] * xcv;
      acc += st[s] * Cm[s];
    }
    const float zv = xz[r * (2 * DIN) + DIN + d];
    y[r * DIN + d] = (acc + Dd * xcv) * silu_f(zv);
  }
}

// --------- attention: one wave per (seq, head); one lane per query ----------
// qkv: [nseq*L, 3*Dm] laid out q|k|v.  o: [nseq*L, Dm].  Requires L <= 32.
__global__ void attention_kernel(const float* __restrict__ qkv,
                                 float* __restrict__ o,
                                 int nseq, int L, int Dm, int nhead)
{
  const int wv = blockIdx.x * (blockDim.x >> 5) + (threadIdx.x >> 5);
  if (wv >= nseq * nhead) return;
  const int hh = wv % nhead;
  const int n  = wv / nhead;
  const int dh = Dm / nhead;
  const int lane = threadIdx.x & 31;
  const int qi = lane < L ? lane : (L - 1);          // clamp for inactive lanes
  const float scale = rsqrtf((float)dh);
  const float* base = qkv + (size_t)n * L * 3 * Dm;
  const float* qp = base + (size_t)qi * 3 * Dm + hh * dh;

  float sc[32];
  float mx = -3.0e38f;
#pragma unroll
  for (int j = 0; j < 32; ++j) {
    float s = 0.f;
    if (j < L) {
      const float* kp = base + (size_t)j * 3 * Dm + Dm + hh * dh;
      for (int dd = 0; dd < dh; ++dd) s += qp[dd] * kp[dd];
      s *= scale;
      mx = fmaxf(mx, s);
    }
    sc[j] = s;
  }
  float den = 0.f;
#pragma unroll
  for (int j = 0; j < 32; ++j)
    if (j < L) { sc[j] = __expf(sc[j] - mx); den += sc[j]; }
  const float inv = 1.f / den;

  if (lane < L) {
    float* op = o + ((size_t)(n * L + lane)) * Dm + hh * dh;
    for (int dd = 0; dd < dh; ++dd) {
      float acc = 0.f;
#pragma unroll
      for (int j = 0; j < 32; ++j)
        if (j < L) acc += sc[j] * base[(size_t)j * 3 * Dm + 2 * Dm + hh * dh + dd];
      op[dd] = acc * inv;
    }
  }
}

// ------------------------------ host helpers --------------------------------
static inline void gemm(hipStream_t s, const float* A, int lda,
                        const float* W, int ldw, const float* bias,
                        const float* res, int ldres,
                        float* C, int ldc, int M, int N, int K, int act)
{
  const int tiles = ((M + 15) / 16) * ((N + 15) / 16);
  dim3 g((tiles + 7) / 8), b(256);                   // 8 waves / block
  if (res)
    gemm_wmma_kernel<0, true><<<g, b, 0, s>>>(A, lda, W, ldw, bias, res, ldres, C, ldc, M, N, K);
  else if (act == 1)
    gemm_wmma_kernel<1, false><<<g, b, 0, s>>>(A, lda, W, ldw, bias, nullptr, 0, C, ldc, M, N, K);
  else if (act == 2)
    gemm_wmma_kernel<2, false><<<g, b, 0, s>>>(A, lda, W, ldw, bias, nullptr, 0, C, ldc, M, N, K);
  else
    gemm_wmma_kernel<0, false><<<g, b, 0, s>>>(A, lda, W, ldw, bias, nullptr, 0, C, ldc, M, N, K);
}

extern "C" void kernel_launch(void* const* d_in, const int* in_sizes, int n_in,
                              void* d_out, int out_size, void* d_ws, size_t ws_size,
                              hipStream_t stream)
{
  (void)in_sizes; (void)n_in; (void)out_size; (void)ws_size;
  const float* x      = (const float*)d_in[0];
  const float* p_in_W = (const float*)d_in[1];
  const float* p_in_b = (const float*)d_in[2];
  const float* m_nw   = (const float*)d_in[3];
  const float* m_inW  = (const float*)d_in[4];
  const float* m_cW   = (const float*)d_in[5];
  const float* m_cb   = (const float*)d_in[6];
  const float* m_xW   = (const float*)d_in[7];
  const float* m_dtW  = (const float*)d_in[8];
  const float* m_dtb  = (const float*)d_in[9];
  const float* m_Alog = (const float*)d_in[10];
  const float* m_D    = (const float*)d_in[11];
  const float* m_oW   = (const float*)d_in[12];
  const float* t_ln1w = (const float*)d_in[13];
  const float* t_ln1b = (const float*)d_in[14];
  const float* t_Win  = (const float*)d_in[15];
  const float* t_bin  = (const float*)d_in[16];
  const float* t_Wo   = (const float*)d_in[17];
  const float* t_bo   = (const float*)d_in[18];
  const float* t_ln2w = (const float*)d_in[19];
  const float* t_ln2b = (const float*)d_in[20];
  const float* t_W1   = (const float*)d_in[21];
  const float* t_b1   = (const float*)d_in[22];
  const float* t_W2   = (const float*)d_in[23];
  const float* t_b2   = (const float*)d_in[24];
  const float* f_nw   = (const float*)d_in[25];
  const float* head_W = (const float*)d_in[26];
  const float* head_b = (const float*)d_in[27];
  const float* e_ln1w = (const float*)d_in[28];
  const float* e_ln1b = (const float*)d_in[29];
  const float* e_Win  = (const float*)d_in[30];
  const float* e_bin  = (const float*)d_in[31];
  const float* e_Wo   = (const float*)d_in[32];
  const float* e_bo   = (const float*)d_in[33];
  const float* e_ln2w = (const float*)d_in[34];
  const float* e_ln2b = (const float*)d_in[35];
  const float* e_W1   = (const float*)d_in[36];
  const float* e_b1   = (const float*)d_in[37];
  const float* e_W2   = (const float*)d_in[38];
  const float* e_b2   = (const float*)d_in[39];
  const float* cls_W  = (const float*)d_in[40];
  const float* cls_b  = (const float*)d_in[41];

  // workspace layout (floats)
  float* ws    = (float*)d_ws;
  float* h     = ws;                                 // RR*DD   trunk activations
  float* a     = h     + (size_t)RR * DD;            // RR*DD   normed acts / feat
  float* big   = a     + (size_t)RR * DD;            // RR*800  xz / qkv / mlp
  float* xc    = big   + (size_t)RR * 800;           // RR*400  conv out / attn out
  float* dblb  = xc    + (size_t)RR * DIN;           // RR*45   dt|B|C
  float* delta = dblb  + (size_t)RR * (DTR + 2*DST); // RR*400
  float* yb    = delta + (size_t)RR * DIN;           // RR*400
  float* f2    = yb    + (size_t)RR * DIN;           // 40*512  pooled features
  float* sa    = f2    + (size_t)40 * 512;           // 40*512
  float* sqkv  = sa    + (size_t)40 * 512;           // 40*1536
  float* so    = sqkv  + (size_t)40 * 1536;          // 40*512
  float* smlp  = so    + (size_t)40 * 512;           // 40*2048

  // input projection: h = x @ p_in_W^T + b   (x viewed as [7200,200])
  gemm(stream, x, DD, p_in_W, DD, p_in_b, nullptr, 0, h, DD, RR, DD, DD, 0);

  const char types[13] = "mmmtttmmmttt";
  int mi = 0, ti = 0;
  for (int li = 0; li < 12; ++li) {
    if (types[li] == 'm') {
      rmsnorm_kernel<<<dim3((RR + 7) / 8), dim3(256), 0, stream>>>(
          h, m_nw + (size_t)mi * DD, a, RR, DD);
      // xz = hn @ inW^T   [RR,800]
      gemm(stream, a, DD, m_inW + (size_t)mi * 2 * DIN * DD, DD,
           nullptr, nullptr, 0, big, 2 * DIN, RR, 2 * DIN, DD, 0);
      { const int total = NSEQ * TT * DIN;
        mamba_conv_kernel<<<dim3((total + 255) / 256), dim3(256), 0, stream>>>(
            big, m_cW + (size_t)mi * DIN * 4, m_cb + (size_t)mi * DIN, xc, NSEQ, TT); }
      // dbl = xc @ xW^T  [RR,45]
      gemm(stream, xc, DIN, m_xW + (size_t)mi * (DTR + 2 * DST) * DIN, DIN,
           nullptr, nullptr, 0, dblb, DTR + 2 * DST, RR, DTR + 2 * DST, DIN, 0);
      // delta = softplus(dt @ dtW^T + dtb)  (dt = dbl[:, :13], lda=45)
      gemm(stream, dblb, DTR + 2 * DST, m_dtW + (size_t)mi * DIN * DTR, DTR,
           m_dtb + (size_t)mi * DIN, nullptr, 0, delta, DIN, RR, DIN, DTR, 2);
      { const int total = NSEQ * DIN;
        mamba_scan_kernel<<<dim3((total + 255) / 256), dim3(256), 0, stream>>>(
            xc, delta, dblb, m_Alog + (size_t)mi * DIN * DST,
            m_D + (size_t)mi * DIN, big, yb, NSEQ, TT); }
      // h = h + y @ oW^T  (in-place residual: each element read before write)
      gemm(stream, yb, DIN, m_oW + (size_t)mi * DD * DIN, DIN,
           nullptr, h, DD, h, DD, RR, DD, DIN, 0);
      ++mi;
    } else {
      layernorm_kernel<<<dim3((RR + 7) / 8), dim3(256), 0, stream>>>(
          h, t_ln1w + (size_t)ti * DD, t_ln1b + (size_t)ti * DD, a, RR, DD);
      // qkv = a @ Win^T + bin  [RR,600]
      gemm(stream, a, DD, t_Win + (size_t)ti * 3 * DD * DD, DD,
           t_bin + (size_t)ti * 3 * DD, nullptr, 0, big, 3 * DD, RR, 3 * DD, DD, 0);
      { const int waves = NSEQ * 8;
        attention_kernel<<<dim3((waves + 7) / 8), dim3(256), 0, stream>>>(
            big, xc, NSEQ, TT, DD, 8); }
      // h = h + o @ Wo^T + bo
      gemm(stream, xc, DD, t_Wo + (size_t)ti * DD * DD, DD,
           t_bo + (size_t)ti * DD, h, DD, h, DD, RR, DD, DD, 0);
      layernorm_kernel<<<dim3((RR + 7) / 8), dim3(256), 0, stream>>>(
          h, t_ln2w + (size_t)ti * DD, t_ln2b + (size_t)ti * DD, a, RR, DD);
      // mlp = gelu(a @ W1^T + b1)  [RR,800]
      gemm(stream, a, DD, t_W1 + (size_t)ti * 800 * DD, DD,
           t_b1 + (size_t)ti * 800, nullptr, 0, big, 800, RR, 800, DD, 1);
      // h = h + mlp @ W2^T + b2
      gemm(stream, big, 800, t_W2 + (size_t)ti * DD * 800, 800,
           t_b2 + (size_t)ti * DD, h, DD, h, DD, RR, DD, 800, 0);
      ++ti;
    }
  }

  // final RMS norm -> a; a's memory is exactly feat viewed as [40, 36000]
  rmsnorm_kernel<<<dim3((RR + 7) / 8), dim3(256), 0, stream>>>(h, f_nw, a, RR, DD);

  // feat2 = gelu(feat @ head_W^T + head_b)  [40,512], K=36000
  gemm(stream, a, CH * TT * DD, head_W, CH * TT * DD, head_b,
       nullptr, 0, f2, 512, BZ * SL, 512, CH * TT * DD, 1);

  // sequence-level transformer block (B=2, L=20, Dm=512, 4 heads)
  layernorm_kernel<<<dim3(5), dim3(256), 0, stream>>>(f2, e_ln1w, e_ln1b, sa, 40, 512);
  gemm(stream, sa, 512, e_Win, 512, e_bin, nullptr, 0, sqkv, 1536, 40, 1536, 512, 0);
  attention_kernel<<<dim3(1), dim3(256), 0, stream>>>(sqkv, so, BZ, SL, 512, 4);
  gemm(stream, so, 512, e_Wo, 512, e_bo, f2, 512, f2, 512, 40, 512, 512, 0);
  layernorm_kernel<<<dim3(5), dim3(256), 0, stream>>>(f2, e_ln2w, e_ln2b, sa, 40, 512);
  gemm(stream, sa, 512, e_W1, 512, e_b1, nullptr, 0, smlp, 2048, 40, 2048, 512, 1);
  gemm(stream, smlp, 2048, e_W2, 2048, e_b2, f2, 512, f2, 512, 40, 512, 2048, 0);

  // classifier -> d_out [40,5]
  gemm(stream, f2, 512, cls_W, 512, cls_b, nullptr, 0,
       (float*)d_out, 5, 40, 5, 512, 0);
}